// deltaSPHModule_45913200394569
// MI455X (gfx1250) — compile-verified
//
#include <hip/hip_runtime.h>

// ---------------------------------------------------------------------------
// delta-SPH for MI455X (gfx1250, wave32).
// - Edge streams staged into LDS with the CDNA5 Tensor Data Mover
//   (tensor_load_to_lds + s_wait_tensorcnt).
// - Pass A's 9-way segment sum is cast as a one-hot matmul and executed with
//   V_WMMA_F32_16X16X4_F32:  C(span x 16) = S(span x 256) * V(256 x 16),
//   cutting ~95 scan ops/edge to ~8.  Scan fallback kept for span > 32.
// - Passes B/C (2 and 1 accumulators) keep the wave32 segmented scan.
// ---------------------------------------------------------------------------

#define WAVE 32
#define TB   256
#define VSTR 20          // V-matrix LDS row stride in floats (bank spread, 16B aligned)

// physics constants (from reference)
#define HSUP   0.01f
#define RESTD  1000.0f
#define ALPHA_ 0.01f
#define DELTA_ 0.1f
#define C0_    24.261079942986875f     // 10*sqrt(2*9.81*0.3)
#define EPS_   1.0e-5f                 // h^2 * 0.1
#define KCONST 22281.692032865347f     // 7/(pi*h^2)

typedef unsigned int u32x4 __attribute__((ext_vector_type(4)));
typedef int          i32x8 __attribute__((ext_vector_type(8)));
typedef int          i32x4 __attribute__((ext_vector_type(4)));
typedef float        v2f   __attribute__((ext_vector_type(2)));
typedef float        v8f   __attribute__((ext_vector_type(8)));

// ---------------------------------------------------------------------------
// TDM: 1-D tile of `nelem` 4-byte elements, global -> LDS (6-arg builtin).
// ---------------------------------------------------------------------------
__device__ __forceinline__ void tdm_load_1d(void* ldsPtr, const void* gPtr,
                                            unsigned int nelem)
{
    unsigned int lds = (unsigned int)(uintptr_t)ldsPtr;   // LDS byte offset
    unsigned long long ga = (unsigned long long)(uintptr_t)gPtr;
    u32x4 g0;
    g0[0] = 1u;                                            // count=1, user mode
    g0[1] = lds;                                           // lds_addr
    g0[2] = (unsigned int)ga;                              // global_addr[31:0]
    g0[3] = ((unsigned int)(ga >> 32) & 0x01FFFFFFu)       // global_addr[56:32]
            | 0x80000000u;                                 // type=2 (bits 127:126)
    i32x8 g1;
    g1[0] = 0x20000;                                       // data_size=2 (4B), mask=0
    g1[1] = (int)((nelem & 0xFFFFu) << 16);                // tensor_dim0[15:0]
    g1[2] = (int)(((nelem >> 16) & 0xFFFFu) | (1u << 16)); // dim0[31:16] | tensor_dim1=1
    g1[3] = (int)(nelem << 16);                            // tile_dim0
    g1[4] = 1;                                             // tile_dim1=1, tile_dim2=0
    g1[5] = (int)nelem;                                    // tensor_dim0_stride
    g1[6] = 0;
    g1[7] = 0;
    i32x4 z4 = {0, 0, 0, 0};
    i32x8 z8 = {0, 0, 0, 0, 0, 0, 0, 0};
    __builtin_amdgcn_tensor_load_to_lds(g0, g1, z4, z4, z8, 0);
}

__device__ __forceinline__ void stage_edges(
    int* s_i, int* s_j, float* s_q, float2* s_dir,
    const int* gi, const int* gj, const float* radial, const float2* dirs,
    int base, int rem)
{
    if ((threadIdx.x >> 5) == 0) {
        tdm_load_1d(s_i,   gi + base,     (unsigned int)rem);
        tdm_load_1d(s_j,   gj + base,     (unsigned int)rem);
        tdm_load_1d(s_q,   radial + base, (unsigned int)rem);
        tdm_load_1d(s_dir, dirs + base,   (unsigned int)(2 * rem));
        __builtin_amdgcn_s_wait_tensorcnt(0);
    }
    __syncthreads();
}

__device__ __forceinline__ float2 wendlandGrad(float q, float dx, float dy) {
    q = fminf(fmaxf(q, 0.0f), 1.0f);
    float om = 1.0f - q;
    float dwdq = -20.0f * KCONST * q * om * om * om;
    float s = dwdq * (1.0f / HSUP);
    return make_float2(s * dx, s * dy);
}

// Wave32 segmented inclusive scan (keys sorted within wave).
__device__ __forceinline__ float segScan(float v, const bool sameUp[5]) {
#pragma unroll
    for (int s = 0; s < 5; ++s) {
        float u = __shfl_up(v, 1 << s, WAVE);
        if (sameUp[s]) v += u;
    }
    return v;
}

__device__ __forceinline__ void makeSameUp(int key, int lane, bool sameUp[5]) {
#pragma unroll
    for (int s = 0; s < 5; ++s) {
        int d = 1 << s;
        int kU = __shfl_up(key, d, WAVE);
        sameUp[s] = (lane >= d) && (kU == key);
    }
}

// ---------------------------------------------------------------------------
// Pass A: Nmat(4) | div(1) | pacc(2) | vdiff(2) via WMMA one-hot reduction.
// ---------------------------------------------------------------------------
__global__ __launch_bounds__(TB) void edgePassA(
    const int* __restrict__ gi, const int* __restrict__ gj,
    const float2* __restrict__ pos, const float2* __restrict__ vel,
    const float* __restrict__ dens, const float* __restrict__ area,
    const float2* __restrict__ dirs, const float* __restrict__ radial,
    float* __restrict__ Nm, float* __restrict__ accDiv,
    float* __restrict__ accPacc, float* __restrict__ accVd, int E)
{
    __shared__ int    s_i[TB];
    __shared__ int    s_j[TB];
    __shared__ float  s_q[TB];
    __shared__ float2 s_dir[TB];
    __shared__ float  Vm[TB * VSTR];   // 256 x 16 value matrix (padded rows)
    __shared__ float  Cred[32 * 16];   // cross-wave C reduction tile

    // zero the C reduction tile (covered by the stage barrier)
    for (int z = threadIdx.x; z < 32 * 16; z += TB) Cred[z] = 0.0f;

    int base = blockIdx.x * TB;
    int rem  = min(TB, E - base);
    stage_edges(s_i, s_j, s_q, s_dir, gi, gj, radial, dirs, base, rem);

    int t = threadIdx.x;
    bool ok = t < rem;
    int tt = ok ? t : 0;

    int key = ok ? s_i[tt] : 0x7FFFFFFF;
    int pj  = s_j[tt];
    float2 dir = s_dir[tt];
    float  q   = s_q[tt];
    int gk = ok ? key : s_i[0];
    float2 Pi = pos[gk], Pj = pos[pj];
    float2 Ui = vel[gk], Uj = vel[pj];
    float di = dens[gk], dj = dens[pj];
    float Vj = area[pj] / dj;

    float rhoi = di * RESTD, rhoj = dj * RESTD;
    float2 gw = wendlandGrad(q, dir.x, dir.y);
    float rjix = Pj.x - Pi.x, rjiy = Pj.y - Pi.y;
    float ujix = Uj.x - Ui.x, ujiy = Uj.y - Ui.y;
    float rji2 = rjix * rjix + rjiy * rjiy + EPS_;

    float n00 = Vj * rjix * gw.x;
    float n01 = Vj * rjix * gw.y;
    float n10 = Vj * rjiy * gw.x;
    float n11 = Vj * rjiy * gw.y;
    float divc = (ujix * gw.x + ujiy * gw.y) * Vj * rhoj;
    float pij = (C0_ * C0_) * ((rhoi - RESTD) + (rhoj - RESTD));
    float pc  = pij * Vj;
    float pax = pc * gw.x, pay = pc * gw.y;
    float piij = fminf(ujix * rjix + ujiy * rjiy, 0.0f) / rji2;
    float vc   = piij * Vj * rhoj / (rhoi + rhoj);
    float vdx = vc * gw.x, vdy = vc * gw.y;

    if (!ok) { n00 = n01 = n10 = n11 = divc = pax = pay = vdx = vdy = 0.0f; }

    // write the value row (cols 0..8 live, 9..15 zero)
    {
        float4* row = (float4*)&Vm[t * VSTR];
        row[0] = make_float4(n00, n01, n10, n11);
        row[1] = make_float4(divc, pax, pay, vdx);
        row[2] = make_float4(vdy, 0.0f, 0.0f, 0.0f);
        row[3] = make_float4(0.0f, 0.0f, 0.0f, 0.0f);
    }
    __syncthreads();

    int baseKey = s_i[0];
    int span    = s_i[rem - 1] - baseKey + 1;   // block-uniform

    if (span <= 32) {
        // ---- WMMA one-hot reduction: C = S(span x 256) * V(256 x 16) ----
        int lane = threadIdx.x & (WAVE - 1);
        int waveId = threadIdx.x >> 5;
        int half = lane >> 4;          // 0: K 0/1, 1: K 2/3 (A 16x4 layout)
        int n    = lane & 15;          // A: row M; B: column N
        v8f c1 = {0.f, 0.f, 0.f, 0.f, 0.f, 0.f, 0.f, 0.f};
        v8f c2 = {0.f, 0.f, 0.f, 0.f, 0.f, 0.f, 0.f, 0.f};
        bool twoTiles = span > 16;
#pragma unroll
        for (int s = 0; s < 8; ++s) {
            int kb = (waveId << 5) + (s << 2);
            int k0 = kb + 2 * half;          // K for VGPR0 of this lane half
            int k1 = k0 + 1;                 // K for VGPR1
            v2f B;
            B.x = Vm[k0 * VSTR + n];
            B.y = Vm[k1 * VSTR + n];
            int key0 = s_i[k0];
            int key1 = s_i[k1];
            v2f A1;
            A1.x = (key0 == baseKey + n) ? 1.0f : 0.0f;
            A1.y = (key1 == baseKey + n) ? 1.0f : 0.0f;
            c1 = __builtin_amdgcn_wmma_f32_16x16x4_f32(
                     false, A1, false, B, (short)0, c1, false, false);
            if (twoTiles) {
                v2f A2;
                A2.x = (key0 == baseKey + 16 + n) ? 1.0f : 0.0f;
                A2.y = (key1 == baseKey + 16 + n) ? 1.0f : 0.0f;
                c2 = __builtin_amdgcn_wmma_f32_16x16x4_f32(
                         false, A2, false, B, (short)0, c2, false, false);
            }
        }
        // combine the 8 wave partials in LDS (C layout: VGPR r holds
        // M = r + 8*half, N = lane&15)
#pragma unroll
        for (int r = 0; r < 8; ++r) {
            atomicAdd(&Cred[(r + 8 * half) * 16 + n], c1[r]);
            if (twoTiles)
                atomicAdd(&Cred[(16 + r + 8 * half) * 16 + n], c2[r]);
        }
        __syncthreads();
        // flush rows 0..span-1, cols 0..8 with global atomics
        for (int t2 = threadIdx.x; t2 < 16 * span; t2 += TB) {
            int m = t2 >> 4, c = t2 & 15;
            if (c < 9) {
                float val = Cred[t2];
                int km = baseKey + m;
                if (c < 4)       atomicAdd(&Nm[4 * km + c], val);
                else if (c == 4) atomicAdd(&accDiv[km], val);
                else if (c < 7)  atomicAdd(&accPacc[2 * km + (c - 5)], val);
                else             atomicAdd(&accVd[2 * km + (c - 7)], val);
            }
        }
    } else {
        // ---- fallback: wave32 segmented scan + tail atomics ----
        int lane = threadIdx.x & (WAVE - 1);
        bool sameUp[5];
        makeSameUp(key, lane, sameUp);
        n00 = segScan(n00, sameUp); n01 = segScan(n01, sameUp);
        n10 = segScan(n10, sameUp); n11 = segScan(n11, sameUp);
        divc = segScan(divc, sameUp);
        pax = segScan(pax, sameUp); pay = segScan(pay, sameUp);
        vdx = segScan(vdx, sameUp); vdy = segScan(vdy, sameUp);
        int kD = __shfl_down(key, 1, WAVE);
        bool tail = (lane == WAVE - 1) || (kD != key);
        if (tail && ok) {
            atomicAdd(&Nm[4 * key + 0], n00);
            atomicAdd(&Nm[4 * key + 1], n01);
            atomicAdd(&Nm[4 * key + 2], n10);
            atomicAdd(&Nm[4 * key + 3], n11);
            atomicAdd(&accDiv[key], divc);
            atomicAdd(&accPacc[2 * key + 0], pax);
            atomicAdd(&accPacc[2 * key + 1], pay);
            atomicAdd(&accVd[2 * key + 0], vdx);
            atomicAdd(&accVd[2 * key + 1], vdy);
        }
    }
}

// ---------------------------------------------------------------------------
// Per-particle 2x2 pseudo-inverse, in place (Nmat -> L)
// ---------------------------------------------------------------------------
__global__ __launch_bounds__(TB) void particlePinv(float* __restrict__ Nm, int N)
{
    int n = blockIdx.x * blockDim.x + threadIdx.x;
    if (n >= N) return;
    float a = Nm[4 * n + 0], b = Nm[4 * n + 1];
    float c = Nm[4 * n + 2], d = Nm[4 * n + 3];
    float det = a * d - b * c;
    float nf  = a * a + b * b + c * c + d * d;
    float l00, l01, l10, l11;
    if (fabsf(det) > 1.0e-6f * nf) {
        float inv = 1.0f / det;
        l00 =  d * inv; l01 = -b * inv;
        l10 = -c * inv; l11 =  a * inv;
    } else if (nf > 1.0e-30f) {
        float inv = 1.0f / nf;
        l00 = a * inv; l01 = c * inv;
        l10 = b * inv; l11 = d * inv;
    } else {
        l00 = l01 = l10 = l11 = 0.0f;
    }
    Nm[4 * n + 0] = l00; Nm[4 * n + 1] = l01;
    Nm[4 * n + 2] = l10; Nm[4 * n + 3] = l11;
}

// ---------------------------------------------------------------------------
// Pass B: gradRho = -seg( 2*(rhoj-rhoi)*Vj * (L[i] @ gradW) )
// ---------------------------------------------------------------------------
__global__ __launch_bounds__(TB) void edgePassB(
    const int* __restrict__ gi, const int* __restrict__ gj,
    const float* __restrict__ dens, const float* __restrict__ area,
    const float2* __restrict__ dirs, const float* __restrict__ radial,
    const float4* __restrict__ Lm, float* __restrict__ gradRho, int E)
{
    __shared__ int    s_i[TB];
    __shared__ int    s_j[TB];
    __shared__ float  s_q[TB];
    __shared__ float2 s_dir[TB];

    int base = blockIdx.x * TB;
    int rem  = min(TB, E - base);
    stage_edges(s_i, s_j, s_q, s_dir, gi, gj, radial, dirs, base, rem);

    int t = threadIdx.x;
    bool ok = t < rem;
    int tt = ok ? t : 0;

    int key = ok ? s_i[tt] : 0x7FFFFFFF;
    int pj  = s_j[tt];
    float2 dir = s_dir[tt];
    float  q   = s_q[tt];
    int gk = ok ? key : s_i[0];
    float di = dens[gk], dj = dens[pj];
    float Vj = area[pj] / dj;

    float2 gw = wendlandGrad(q, dir.x, dir.y);
    float4 L = Lm[gk];
    float gx = L.x * gw.x + L.y * gw.y;
    float gy = L.z * gw.x + L.w * gw.y;
    float coef = -2.0f * (RESTD * (dj - di)) * Vj;
    float vx = coef * gx, vy = coef * gy;
    if (!ok) { vx = vy = 0.0f; }

    int lane = threadIdx.x & (WAVE - 1);
    bool sameUp[5];
    makeSameUp(key, lane, sameUp);
    vx = segScan(vx, sameUp);
    vy = segScan(vy, sameUp);

    int kD = __shfl_down(key, 1, WAVE);
    bool tail = (lane == WAVE - 1) || (kD != key);
    if (tail && ok) {
        atomicAdd(&gradRho[2 * key + 0], vx);
        atomicAdd(&gradRho[2 * key + 1], vy);
    }
}

// ---------------------------------------------------------------------------
// Pass C: density diffusion  seg( (psi . gradW) * Vj )
// ---------------------------------------------------------------------------
__global__ __launch_bounds__(TB) void edgePassC(
    const int* __restrict__ gi, const int* __restrict__ gj,
    const float2* __restrict__ pos,
    const float* __restrict__ dens, const float* __restrict__ area,
    const float2* __restrict__ dirs, const float* __restrict__ radial,
    const float2* __restrict__ gradRho, float* __restrict__ accDD, int E)
{
    __shared__ int    s_i[TB];
    __shared__ int    s_j[TB];
    __shared__ float  s_q[TB];
    __shared__ float2 s_dir[TB];

    int base = blockIdx.x * TB;
    int rem  = min(TB, E - base);
    stage_edges(s_i, s_j, s_q, s_dir, gi, gj, radial, dirs, base, rem);

    int t = threadIdx.x;
    bool ok = t < rem;
    int tt = ok ? t : 0;

    int key = ok ? s_i[tt] : 0x7FFFFFFF;
    int pj  = s_j[tt];
    float2 dir = s_dir[tt];
    float  q   = s_q[tt];
    int gk = ok ? key : s_i[0];
    float2 Pi = pos[gk], Pj = pos[pj];
    float di = dens[gk], dj = dens[pj];
    float Vj = area[pj] / dj;

    float rjix = Pj.x - Pi.x, rjiy = Pj.y - Pi.y;
    float rji2 = rjix * rjix + rjiy * rjiy + EPS_;
    float2 gw = wendlandGrad(q, dir.x, dir.y);
    float2 gRi = gradRho[gk], gRj = gradRho[pj];

    float s = 2.0f * (RESTD * (dj - di)) / rji2;
    float psix = s * rjix - (gRi.x + gRj.x);
    float psiy = s * rjiy - (gRi.y + gRj.y);
    float ddc = (psix * gw.x + psiy * gw.y) * Vj;
    if (!ok) ddc = 0.0f;

    int lane = threadIdx.x & (WAVE - 1);
    bool sameUp[5];
    makeSameUp(key, lane, sameUp);
    ddc = segScan(ddc, sameUp);

    int kD = __shfl_down(key, 1, WAVE);
    bool tail = (lane == WAVE - 1) || (kD != key);
    if (tail && ok) atomicAdd(&accDD[key], ddc);
}

// ---------------------------------------------------------------------------
// Finalize: pack (N,6) output
// ---------------------------------------------------------------------------
__global__ __launch_bounds__(TB) void finalizeK(
    const float* __restrict__ dens,
    const float* __restrict__ accDiv, const float* __restrict__ accPacc,
    const float* __restrict__ accVd,  const float* __restrict__ accDD,
    float* __restrict__ out, int N)
{
    int n = blockIdx.x * blockDim.x + threadIdx.x;
    if (n >= N) return;
    float rho  = dens[n] * RESTD;
    float dpdt = -accDiv[n] + (HSUP * DELTA_ * C0_) * accDD[n];
    float pres = (C0_ * C0_) * (rho - RESTD);
    float inv  = -1.0f / rho;
    out[6 * n + 0] = dpdt;
    out[6 * n + 1] = pres;
    out[6 * n + 2] = inv * accPacc[2 * n + 0];
    out[6 * n + 3] = inv * accPacc[2 * n + 1];
    out[6 * n + 4] = (HSUP * ALPHA_ * C0_) * accVd[2 * n + 0];
    out[6 * n + 5] = (HSUP * ALPHA_ * C0_) * accVd[2 * n + 1];
}

// ---------------------------------------------------------------------------
extern "C" void kernel_launch(void* const* d_in, const int* in_sizes, int n_in,
                              void* d_out, int out_size, void* d_ws, size_t ws_size,
                              hipStream_t stream)
{
    const int*    gi     = (const int*)d_in[0];
    const int*    gj     = (const int*)d_in[1];
    const float2* pos    = (const float2*)d_in[2];
    const float2* vel    = (const float2*)d_in[3];
    const float*  dens   = (const float*)d_in[4];
    const float*  area   = (const float*)d_in[5];
    const float2* dirs   = (const float2*)d_in[6];
    const float*  radial = (const float*)d_in[7];
    const int E = in_sizes[0];
    const int N = in_sizes[4];

    float* ws      = (float*)d_ws;
    float* Nm      = ws;                      // 4N (becomes L in place)
    float* accDiv  = ws + (size_t)4 * N;      // N
    float* accPacc = ws + (size_t)5 * N;      // 2N
    float* accVd   = ws + (size_t)7 * N;      // 2N
    float* gradRho = ws + (size_t)9 * N;      // 2N
    float* accDD   = ws + (size_t)11 * N;     // N   -> total 12N floats

    (void)hipMemsetAsync(d_ws, 0, (size_t)12 * N * sizeof(float), stream);

    int be = (E + TB - 1) / TB;
    int bn = (N + TB - 1) / TB;

    edgePassA<<<be, TB, 0, stream>>>(gi, gj, pos, vel, dens, area, dirs, radial,
                                     Nm, accDiv, accPacc, accVd, E);
    particlePinv<<<bn, TB, 0, stream>>>(Nm, N);
    edgePassB<<<be, TB, 0, stream>>>(gi, gj, dens, area, dirs, radial,
                                     (const float4*)Nm, gradRho, E);
    edgePassC<<<be, TB, 0, stream>>>(gi, gj, pos, dens, area, dirs, radial,
                                     (const float2*)gradRho, accDD, E);
    finalizeK<<<bn, TB, 0, stream>>>(dens, accDiv, accPacc, accVd, accDD,
                                     (float*)d_out, N);
}